// HashGridEmbedder_20057497272434
// MI455X (gfx1250) — compile-verified
//
#include <hip/hip_runtime.h>
#include <hip/hip_bf16.h>

#define HG_LEVELS 16
#define HG_LOG2T  19

namespace {

constexpr unsigned kT      = 1u << HG_LOG2T;
constexpr unsigned kTMask  = kT - 1u;
constexpr int      kBlock  = 256;
constexpr int      kPtsPerBlock = kBlock / HG_LEVELS;   // 16 points per block

// scale[l] = float32(16 * (4096/16)^(l/15) - 1)  (tcnn per-level geometry)
__constant__ float c_scale[HG_LEVELS] = {
    15.0f,         22.1563078f,  32.5134117f,  47.5029301f,
    69.1967981f,   100.5936673f, 146.0333894f, 211.7968995f,
    306.9744140f,  444.7218884f, 644.0795776f, 932.6038400f,
    1350.1761000f, 1954.5156000f, 2829.1576000f, 4095.0f};

// res[l] = ceil(scale)+1; dense (res^3 <= 2^19) only for levels 0..4
__constant__ unsigned c_res[HG_LEVELS] = {
    16u, 24u, 34u, 49u, 71u, 0u, 0u, 0u, 0u, 0u, 0u, 0u, 0u, 0u, 0u, 0u};

__global__ __launch_bounds__(kBlock) void hashgrid_fwd(
    const float* __restrict__ inputs,   // [N,3]
    const float* __restrict__ table,    // [16*T, 2]
    const float* __restrict__ aabb,     // [2,3]
    float* __restrict__ out,            // [N,32]
    int n_points) {
  __shared__ float    s_scale[HG_LEVELS];
  __shared__ unsigned s_res[HG_LEVELS];
  __shared__ float    s_xyz[kPtsPerBlock * 3];   // 48 floats

  const int tid = threadIdx.x;

  if (tid < HG_LEVELS) {
    s_scale[tid] = c_scale[tid];
    s_res[tid]   = c_res[tid];
  }

  // ---- Stage this block's 16 input points into LDS (gfx1250 async path) ----
  const int n3      = n_points * 3;
  const int in_base = blockIdx.x * (kPtsPerBlock * 3);
#if __has_builtin(__builtin_amdgcn_global_load_async_to_lds_b32)
  if (tid < kPtsPerBlock * 3 && in_base + tid < n3) {
    typedef __attribute__((address_space(1))) int* gptr_t;
    typedef __attribute__((address_space(3))) int* lptr_t;
    gptr_t g = (gptr_t)(unsigned long long)(inputs + in_base + tid);
    lptr_t l = (lptr_t)(unsigned)(unsigned long long)(&s_xyz[tid]);
    __builtin_amdgcn_global_load_async_to_lds_b32(g, l, 0, 0);
  }
#if __has_builtin(__builtin_amdgcn_s_wait_asynccnt)
  __builtin_amdgcn_s_wait_asynccnt(0);
#else
  asm volatile("s_wait_asynccnt 0" ::: "memory");
#endif
#else
  if (tid < kPtsPerBlock * 3 && in_base + tid < n3)
    s_xyz[tid] = inputs[in_base + tid];
#endif

  // Warm L2 for the next block's inputs (lowers to global_prefetch_b8).
  if (tid == 0 && in_base + kPtsPerBlock * 3 < n3)
    __builtin_prefetch(inputs + in_base + kPtsPerBlock * 3, 0, 0);

  __syncthreads();

  const int t   = blockIdx.x * kBlock + tid;
  const int lp  = tid >> 4;        // local point index (0..15)
  const int lvl = tid & 15;        // level index       (0..15)
  const int n   = blockIdx.x * kPtsPerBlock + lp;
  if (n >= n_points) return;

  // ---- AABB normalize + clip (matches reference: (x-lo)/(hi-lo), clip) ----
  const float lox = aabb[0], loy = aabb[1], loz = aabb[2];
  const float hix = aabb[3], hiy = aabb[4], hiz = aabb[5];
  float x = s_xyz[lp * 3 + 0];
  float y = s_xyz[lp * 3 + 1];
  float z = s_xyz[lp * 3 + 2];
  x = fminf(fmaxf((x - lox) / (hix - lox), 0.0f), 1.0f);
  y = fminf(fmaxf((y - loy) / (hiy - loy), 0.0f), 1.0f);
  z = fminf(fmaxf((z - loz) / (hiz - loz), 0.0f), 1.0f);

  // ---- Per-level grid coords + trilinear weights ----
  const float    scale = s_scale[lvl];
  const unsigned res   = s_res[lvl];
  const bool     dense = (lvl < 5);

  const float px = x * scale + 0.5f;
  const float py = y * scale + 0.5f;
  const float pz = z * scale + 0.5f;
  const float fx = floorf(px), fy = floorf(py), fz = floorf(pz);
  const float wx = px - fx,   wy = py - fy,   wz = pz - fz;
  const unsigned ix = (unsigned)fx, iy = (unsigned)fy, iz = (unsigned)fz;

  const unsigned res2 = res * res;
  const float2* __restrict__ tb =
      reinterpret_cast<const float2*>(table) + ((size_t)lvl << HG_LOG2T);

  // ---- Compute all 8 corner indices, issue all 8 b64 gathers, then blend ----
  float2 feat[8];
  float  wgt[8];
#pragma unroll
  for (int c = 0; c < 8; ++c) {
    const unsigned bx = (c >> 2) & 1u, by = (c >> 1) & 1u, bz = c & 1u;
    const unsigned cx = ix + bx, cy = iy + by, cz = iz + bz;
    unsigned h = cx ^ (cy * 2654435761u) ^ (cz * 805459861u);
    h &= kTMask;
    const unsigned d   = cx + cy * res + cz * res2;
    const unsigned idx = dense ? d : h;
    feat[c] = tb[idx];                       // global_load_b64 gather (L2-resident)
    const float a  = bx ? wx : 1.0f - wx;
    const float b  = by ? wy : 1.0f - wy;
    const float cc = bz ? wz : 1.0f - wz;
    wgt[c] = a * b * cc;
  }

  float2 acc;
  acc.x = 0.0f;
  acc.y = 0.0f;
#pragma unroll
  for (int c = 0; c < 8; ++c) {
    acc.x = fmaf(wgt[c], feat[c].x, acc.x);
    acc.y = fmaf(wgt[c], feat[c].y, acc.y);
  }

  // out[n, lvl*2 + {0,1}] -> float2 index n*16+lvl == t : coalesced b64 store
  reinterpret_cast<float2*>(out)[t] = acc;
}

}  // namespace

extern "C" void kernel_launch(void* const* d_in, const int* in_sizes, int n_in,
                              void* d_out, int out_size, void* d_ws, size_t ws_size,
                              hipStream_t stream) {
  const float* inputs = (const float*)d_in[0];
  const float* table  = (const float*)d_in[1];
  const float* aabb   = (const float*)d_in[2];
  float* out          = (float*)d_out;

  const int n_points = in_sizes[0] / 3;
  const int total    = n_points * HG_LEVELS;
  const int blocks   = (total + kBlock - 1) / kBlock;

  hipLaunchKernelGGL(hashgrid_fwd, dim3(blocks), dim3(kBlock), 0, stream,
                     inputs, table, aabb, out, n_points);
}